// SVFDeformer_24988119728531
// MI455X (gfx1250) — compile-verified
//
#include <hip/hip_runtime.h>

// ---------------------------------------------------------------------------
// SVF multi-level trilinear sampler for MI455X (gfx1250, wave32).
//
// Roofline: ~100 MB HBM traffic (48 read + 48 write + 3.4 one-time) ->
// ~4.3 us floor at 23.3 TB/s. Grids (L0 32^3, L1 64^3, 3ch) are L2-resident
// (4.5 MB AoS vs 192 MB L2), so the kernel is gather-issue/latency bound.
// WMMA is structurally inapplicable (per-point gathered B operand).
//
// Strategy:
//  * Repack grids to [D,H,W,4] float4 so each trilinear corner is ONE
//    global_load_b128 carrying all 3 channels.
//  * All 8 corner loads per level share a single VGPR address with
//    compile-time immediate offsets (S is a template parameter).
//  * 2 independent points per thread, straight-line, for ~32 outstanding
//    gathers per lane-context (L2 latency hiding).
//  * Border padding folded into the lerp: base corner clamped to S-2 with
//    fraction relative to the base (f==1 at the edge selects last texel).
// ---------------------------------------------------------------------------

typedef float v4f __attribute__((ext_vector_type(4)));
typedef float v3f __attribute__((ext_vector_type(3)));

struct F3 { float x, y, z; };   // align 4, size 12 -> b96-friendly

__device__ __forceinline__ v3f lerp3(v3f a, v3f b, float t) {
    return a + t * (b - a);
}

// --- AoS-path level sampler: 8 x global_load_b128 from one base address ----
template <int S>
__device__ __forceinline__ v3f sample_aos(const v4f* __restrict__ grid,
                                          float nx, float ny, float nz) {
    const float h = 0.5f * (float)(S - 1);
    // align_corners unnormalize + border clamp (subsumes the +-1.5 clip)
    float cx = fminf(fmaxf(fmaf(nx, h, h), 0.0f), (float)(S - 1));
    float cy = fminf(fmaxf(fmaf(ny, h, h), 0.0f), (float)(S - 1));
    float cz = fminf(fmaxf(fmaf(nz, h, h), 0.0f), (float)(S - 1));
    int xb = min((int)cx, S - 2);
    int yb = min((int)cy, S - 2);
    int zb = min((int)cz, S - 2);
    float fx = cx - (float)xb;
    float fy = cy - (float)yb;
    float fz = cz - (float)zb;

    const v4f* base = grid + (((zb * S) + yb) * S + xb);
    v4f c000 = base[0];             v4f c001 = base[1];
    v4f c010 = base[S];             v4f c011 = base[S + 1];
    v4f c100 = base[S * S];         v4f c101 = base[S * S + 1];
    v4f c110 = base[S * S + S];     v4f c111 = base[S * S + S + 1];

    v3f a0 = lerp3(c000.xyz, c001.xyz, fx);
    v3f a1 = lerp3(c010.xyz, c011.xyz, fx);
    v3f a2 = lerp3(c100.xyz, c101.xyz, fx);
    v3f a3 = lerp3(c110.xyz, c111.xyz, fx);
    v3f b0 = lerp3(a0, a1, fy);
    v3f b1 = lerp3(a2, a3, fy);
    return lerp3(b0, b1, fz);
}

// --- Fallback: gather directly from [C,D,H,W] (no workspace needed) --------
template <int S>
__device__ __forceinline__ v3f sample_cdw(const float* __restrict__ grid,
                                          float nx, float ny, float nz) {
    const float h = 0.5f * (float)(S - 1);
    float cx = fminf(fmaxf(fmaf(nx, h, h), 0.0f), (float)(S - 1));
    float cy = fminf(fmaxf(fmaf(ny, h, h), 0.0f), (float)(S - 1));
    float cz = fminf(fmaxf(fmaf(nz, h, h), 0.0f), (float)(S - 1));
    int xb = min((int)cx, S - 2);
    int yb = min((int)cy, S - 2);
    int zb = min((int)cz, S - 2);
    float fx = cx - (float)xb;
    float fy = cy - (float)yb;
    float fz = cz - (float)zb;

    const float* base = grid + (((zb * S) + yb) * S + xb);
    v3f r;
#pragma unroll
    for (int c = 0; c < 3; ++c) {
        const float* g = base + c * (S * S * S);   // <= 2.1MB, fits IOFFSET
        float c000 = g[0];             float c001 = g[1];
        float c010 = g[S];             float c011 = g[S + 1];
        float c100 = g[S * S];         float c101 = g[S * S + 1];
        float c110 = g[S * S + S];     float c111 = g[S * S + S + 1];
        float a0 = c000 + fx * (c001 - c000);
        float a1 = c010 + fx * (c011 - c010);
        float a2 = c100 + fx * (c101 - c100);
        float a3 = c110 + fx * (c111 - c110);
        float b0 = a0 + fy * (a1 - a0);
        float b1 = a2 + fy * (a3 - a2);
        r[c] = b0 + fz * (b1 - b0);
    }
    return r;
}

// --- Repack [3,S,S,S] -> [S,S,S,4] float4 ----------------------------------
__global__ void __launch_bounds__(256)
svf_repack_kernel(const float* __restrict__ src, v4f* __restrict__ dst, int nvox) {
    int i = blockIdx.x * blockDim.x + threadIdx.x;
    if (i >= nvox) return;
    v4f v;
    v.x = src[i];
    v.y = src[i + nvox];
    v.z = src[i + 2 * nvox];
    v.w = 0.0f;
    dst[i] = v;                       // global_store_b128
}

// --- Main AoS kernel: 2 independent points per thread ----------------------
__global__ void __launch_bounds__(256)
svf_aos_kernel(const float* __restrict__ xw,
               const float* __restrict__ center,
               const float* __restrict__ halfv,
               const v4f* __restrict__ g0,
               const v4f* __restrict__ g1,
               float* __restrict__ out, int G) {
    int t  = blockIdx.x * blockDim.x + threadIdx.x;
    int nA = (G + 1) >> 1;
    if (t >= nA) return;

    int gA  = t;
    int gB  = t + nA;
    int gBc = min(gB, G - 1);          // clamped for safe speculative loads

    // Uniform transform: n = p*inv + bias, with inv/bias computed once.
    float ix = 1.0f / (halfv[0] + 1e-8f);
    float iy = 1.0f / (halfv[1] + 1e-8f);
    float iz = 1.0f / (halfv[2] + 1e-8f);
    float bx = -center[0] * ix;
    float by = -center[1] * iy;
    float bz = -center[2] * iz;

    F3 pA = ((const F3*)xw)[gA];       // b96 streaming loads
    F3 pB = ((const F3*)xw)[gBc];

    float nxA = fmaf(pA.x, ix, bx), nyA = fmaf(pA.y, iy, by), nzA = fmaf(pA.z, iz, bz);
    float nxB = fmaf(pB.x, ix, bx), nyB = fmaf(pB.y, iy, by), nzB = fmaf(pB.z, iz, bz);

    // Interleave the two independent gather trees so ~32 b128 loads can be
    // in flight per lane-context.
    v3f rA = sample_aos<32>(g0, nxA, nyA, nzA);
    v3f rB = sample_aos<32>(g0, nxB, nyB, nzB);
    rA += sample_aos<64>(g1, nxA, nyA, nzA);
    rB += sample_aos<64>(g1, nxB, nyB, nzB);

    F3 oA; oA.x = rA.x; oA.y = rA.y; oA.z = rA.z;
    ((F3*)out)[gA] = oA;               // b96 streaming store
    if (gB < G) {
        F3 oB; oB.x = rB.x; oB.y = rB.y; oB.z = rB.z;
        ((F3*)out)[gB] = oB;
    }
}

// --- Fallback kernel (original grid layout, 1 point per thread) ------------
__global__ void __launch_bounds__(256)
svf_cdw_kernel(const float* __restrict__ xw,
               const float* __restrict__ center,
               const float* __restrict__ halfv,
               const float* __restrict__ L0,
               const float* __restrict__ L1,
               float* __restrict__ out, int G) {
    int g = blockIdx.x * blockDim.x + threadIdx.x;
    if (g >= G) return;

    float ix = 1.0f / (halfv[0] + 1e-8f);
    float iy = 1.0f / (halfv[1] + 1e-8f);
    float iz = 1.0f / (halfv[2] + 1e-8f);
    float bx = -center[0] * ix;
    float by = -center[1] * iy;
    float bz = -center[2] * iz;

    F3 p = ((const F3*)xw)[g];
    float nx = fmaf(p.x, ix, bx), ny = fmaf(p.y, iy, by), nz = fmaf(p.z, iz, bz);

    v3f r = sample_cdw<32>(L0, nx, ny, nz) + sample_cdw<64>(L1, nx, ny, nz);

    F3 o; o.x = r.x; o.y = r.y; o.z = r.z;
    ((F3*)out)[g] = o;
}

// ---------------------------------------------------------------------------
extern "C" void kernel_launch(void* const* d_in, const int* in_sizes, int n_in,
                              void* d_out, int out_size, void* d_ws, size_t ws_size,
                              hipStream_t stream) {
    const float* xw     = (const float*)d_in[0];   // [G,3]
    const float* center = (const float*)d_in[1];   // [3]
    const float* halfv  = (const float*)d_in[2];   // [3]
    const float* L0     = (const float*)d_in[3];   // [3,32,32,32]
    const float* L1     = (const float*)d_in[4];   // [3,64,64,64]
    float* out          = (float*)d_out;           // [G,3]

    const int G  = in_sizes[0] / 3;
    const int n0 = 32 * 32 * 32;
    const int n1 = 64 * 64 * 64;
    const size_t need = (size_t)(n0 + n1) * sizeof(v4f);   // ~4.5 MB

    const int BS = 256;
    if (ws_size >= need) {
        v4f* a0 = (v4f*)d_ws;
        v4f* a1 = a0 + n0;
        svf_repack_kernel<<<(n0 + BS - 1) / BS, BS, 0, stream>>>(L0, a0, n0);
        svf_repack_kernel<<<(n1 + BS - 1) / BS, BS, 0, stream>>>(L1, a1, n1);
        const int nThreads = (G + 1) >> 1;         // 2 points per thread
        svf_aos_kernel<<<(nThreads + BS - 1) / BS, BS, 0, stream>>>(
            xw, center, halfv, a0, a1, out, G);
    } else {
        svf_cdw_kernel<<<(G + BS - 1) / BS, BS, 0, stream>>>(
            xw, center, halfv, L0, L1, out, G);
    }
}